// CharacteristicLineEncoder_335007449160
// MI455X (gfx1250) — compile-verified
//
#include <hip/hip_runtime.h>

// CDNA5 (gfx1250) fused CharacteristicLineEncoder.
// One workgroup (128 threads = 4 wave32) per batch element.
// Embedding GEMMs ([128,3]x[3,300], K padded 3->4) run on V_WMMA_F32_16X16X4_F32.
// Memory-bound: ~140 MB moved (x in, out out) vs 2.5 GFLOP -> fusion is the win
// (avoids 2 x 629 MB [B,E,L] intermediates).

typedef __attribute__((ext_vector_type(2))) float v2f;
typedef __attribute__((ext_vector_type(8))) float v8f;

namespace {
constexpr int kDIM = 3;
constexpr int kL   = 300;
constexpr int kS   = 60;
constexpr int kE   = 128;
constexpr int kP   = 5;
constexpr int kXS  = 304;   // L padded to 19 column-tiles of 16
constexpr int kLS  = 84;    // LDS row stride (words) for le chunk: 336 B -> 16B-aligned rows for b128
}

// Sum over each 16-lane half of a wave32 via ds_swizzle XOR mode (no bounds-check cruft).
__device__ __forceinline__ float half16_sum(float v) {
  v += __int_as_float(__builtin_amdgcn_ds_swizzle(__float_as_int(v), 0x041f)); // xor 1
  v += __int_as_float(__builtin_amdgcn_ds_swizzle(__float_as_int(v), 0x081f)); // xor 2
  v += __int_as_float(__builtin_amdgcn_ds_swizzle(__float_as_int(v), 0x101f)); // xor 4
  v += __int_as_float(__builtin_amdgcn_ds_swizzle(__float_as_int(v), 0x201f)); // xor 8
  return v;
}

__global__ __launch_bounds__(128)
void cle_fused_kernel(const float* __restrict__ x,
                      const float* __restrict__ W_le, const float* __restrict__ b_le,
                      const float* __restrict__ W_lr, const float* __restrict__ b_lr,
                      const float* __restrict__ W_ge, const float* __restrict__ b_ge,
                      const float* __restrict__ W_gn, const float* __restrict__ b_gn,
                      float* __restrict__ out)
{
  __shared__ float xs[4 * kXS];      // rows 0..2: x[b] (cols 300..303 zero); row 3: all zeros (K=3 pad)
  __shared__ float wgn_s[kXS];       // W_gn, zero padded
  __shared__ float ldc[kE * kLS];    // relu(le) chunk: 128 e x 80 l (stride 84 words)
  __shared__ float gf_s[kE];         // global feature
  __shared__ float wlr_s[kS * kP];
  __shared__ float blr_s[kS];

  const int tid  = threadIdx.x;
  const int b    = blockIdx.x;
  const int lane = tid & 31;
  const int wv   = tid >> 5;
  const int n    = lane & 15;   // tile row (A) / tile column (B,C,D)
  const int hi   = lane >> 4;   // half-wave select

  // ---- stage x[b] + small tables into LDS ----
  const float* xb = x + (size_t)b * (kDIM * kL);
  for (int j = tid; j < 4 * kXS; j += 128) {
    int c = j / kXS, l = j - c * kXS;
    xs[j] = (c < kDIM && l < kL) ? xb[c * kL + l] : 0.0f;
  }
  for (int j = tid; j < kXS; j += 128) wgn_s[j] = (j < kL) ? W_gn[j] : 0.0f;
  for (int j = tid; j < kS * kP; j += 128) wlr_s[j] = W_lr[j];
  if (tid < kS) blr_s[tid] = b_lr[tid];
  __syncthreads();

  // Uniform per-lane LDS offsets for the 4x16 f32 B operand:
  // V0 = K0 (lanes 0-15) | K2 (lanes 16-31); V1 = K1 | K3, with row 3 = zeros.
  const int xoff0 = (hi ? 2 : 0) * kXS + n;
  const int xoff1 = (hi ? 3 : 1) * kXS + n;

  // ---- per-wave constants: this wave's two 16-row e-tiles ----
  const int eb0 = 32 * wv;
  const int eb1 = 32 * wv + 16;

  // 16x4 f32 A layout: VGPR0 = K0 | K2; VGPR1 = K1 | K3(=0).
  const int c02 = hi ? 2 : 0;
  const int e0  = eb0 + n, e1 = eb1 + n;

  v2f age0, age1;
  {
    float t0 = W_ge[e0 * 3 + 1], t1 = W_ge[e1 * 3 + 1];
    age0.x = W_ge[e0 * 3 + c02];  age0.y = hi ? 0.0f : t0;
    age1.x = W_ge[e1 * 3 + c02];  age1.y = hi ? 0.0f : t1;
  }
  // C[M][N] = bias[e_base + M], broadcast along N (reg r: M = r + 8*hi)
  v8f cge0, cge1;
#pragma unroll
  for (int r = 0; r < 8; ++r) {
    const int o = r + (hi ? 8 : 0);
    cge0[r] = b_ge[eb0 + o];
    cge1[r] = b_ge[eb1 + o];
  }

  // ---- global branch: ge = relu(W_ge@x + b_ge); gf = relu(sum_l W_gn[l]*ge[:,l] + b_gn) ----
  v8f g0 = {}; v8f g1 = {};
#pragma clang loop unroll(disable)
  for (int lt = 0; lt < 19; ++lt) {
    const int lb = lt * 16;
    v2f bop;
    bop.x = xs[xoff0 + lb];
    bop.y = xs[xoff1 + lb];
    const float wgn = wgn_s[lb + n];   // 0 for padded columns -> masks relu(bias) garbage
    v8f d0 = __builtin_amdgcn_wmma_f32_16x16x4_f32(
        false, age0, false, bop, (short)0, cge0, false, false);
    v8f d1 = __builtin_amdgcn_wmma_f32_16x16x4_f32(
        false, age1, false, bop, (short)0, cge1, false, false);
#pragma unroll
    for (int r = 0; r < 8; ++r) {
      g0[r] += fmaxf(d0[r], 0.0f) * wgn;
      g1[r] += fmaxf(d1[r], 0.0f) * wgn;
    }
  }
  // reduce over the 16 columns held across each 16-lane half
#pragma unroll
  for (int r = 0; r < 8; ++r) {
    g0[r] = half16_sum(g0[r]);
    g1[r] = half16_sum(g1[r]);
  }
  {
    const float bgn = b_gn[0];
    if (lane == 0) {
#pragma unroll
      for (int r = 0; r < 8; ++r) {
        gf_s[eb0 + r] = fmaxf(g0[r] + bgn, 0.0f);
        gf_s[eb1 + r] = fmaxf(g1[r] + bgn, 0.0f);
      }
    } else if (lane == 16) {
#pragma unroll
      for (int r = 0; r < 8; ++r) {
        gf_s[eb0 + 8 + r] = fmaxf(g0[r] + bgn, 0.0f);
        gf_s[eb1 + 8 + r] = fmaxf(g1[r] + bgn, 0.0f);
      }
    }
  }

  // ---- local-branch operands (set up after global pass to limit live ranges) ----
  v2f ale0, ale1;
  {
    float t0 = W_le[e0 * 3 + 1], t1 = W_le[e1 * 3 + 1];
    ale0.x = W_le[e0 * 3 + c02];  ale0.y = hi ? 0.0f : t0;
    ale1.x = W_le[e1 * 3 + c02];  ale1.y = hi ? 0.0f : t1;
  }
  v8f cle0, cle1;
#pragma unroll
  for (int r = 0; r < 8; ++r) {
    const int o = r + (hi ? 8 : 0);
    cle0[r] = b_le[eb0 + o];
    cle1[r] = b_le[eb1 + o];
  }

  __syncthreads();
  const float gfv = gf_s[tid];    // pass-2 thread owns e = tid

  // ---- local branch in 80-column chunks (LCM(16,5)): WMMA -> LDS -> region reduce ----
  for (int ck = 0; ck < 4; ++ck) {
    const int ntiles = (ck < 3) ? 5 : 4;   // chunk 3: l 240..303 (valid to 299)
#pragma clang loop unroll(disable)
    for (int tt = 0; tt < ntiles; ++tt) {
      const int lb = (5 * ck + tt) * 16;
      v2f bop;
      bop.x = xs[xoff0 + lb];
      bop.y = xs[xoff1 + lb];
      v8f d0 = __builtin_amdgcn_wmma_f32_16x16x4_f32(
          false, ale0, false, bop, (short)0, cle0, false, false);
      v8f d1 = __builtin_amdgcn_wmma_f32_16x16x4_f32(
          false, ale1, false, bop, (short)0, cle1, false, false);
      const int col = tt * 16 + n;
#pragma unroll
      for (int r = 0; r < 8; ++r) {
        const int m = r + (hi ? 8 : 0);
        ldc[(eb0 + m) * kLS + col] = fmaxf(d0[r], 0.0f);
        ldc[(eb1 + m) * kLS + col] = fmaxf(d1[r], 0.0f);
      }
    }
    __syncthreads();

    // groups of 4 regions: 5 x ds_load_b128 covers 20 l-values = 4 regions
    const int ngrp = (ck < 3) ? 4 : 3;     // chunk 3: 12 regions
    const float4* row4 = (const float4*)(ldc + tid * kLS);  // 16B-aligned (kLS*4 = 336)
    float* ob = out + (size_t)b * (kS * kE) + (16 * ck) * kE + tid;
    for (int g = 0; g < ngrp; ++g) {
      float4 qv[5];
#pragma unroll
      for (int j = 0; j < 5; ++j) qv[j] = row4[g * 5 + j];
      const float* lv = (const float*)&qv[0];
#pragma unroll
      for (int q = 0; q < 4; ++q) {
        const int s = 16 * ck + 4 * g + q;
        float acc = blr_s[s] + gfv;
#pragma unroll
        for (int p = 0; p < kP; ++p)
          acc = fmaf(wlr_s[s * kP + p], lv[q * 5 + p], acc);
        ob[(4 * g + q) * kE] = acc;        // coalesced b32 stores
      }
    }
    __syncthreads();
  }
}

extern "C" void kernel_launch(void* const* d_in, const int* in_sizes, int n_in,
                              void* d_out, int out_size, void* d_ws, size_t ws_size,
                              hipStream_t stream) {
  (void)in_sizes; (void)n_in; (void)d_ws; (void)ws_size; (void)out_size;
  const float* x    = (const float*)d_in[0];
  const float* W_le = (const float*)d_in[1];
  const float* b_le = (const float*)d_in[2];
  const float* W_lr = (const float*)d_in[3];
  const float* b_lr = (const float*)d_in[4];
  const float* W_ge = (const float*)d_in[5];
  const float* b_ge = (const float*)d_in[6];
  const float* W_gn = (const float*)d_in[7];
  const float* b_gn = (const float*)d_in[8];
  float* out = (float*)d_out;

  cle_fused_kernel<<<4096, 128, 0, stream>>>(x, W_le, b_le, W_lr, b_lr,
                                             W_ge, b_ge, W_gn, b_gn, out);
}